// ViewContrastiveLoss_15710990369515
// MI455X (gfx1250) — compile-verified
//
#include <hip/hip_runtime.h>
#include <hip/hip_bf16.h>

typedef __attribute__((ext_vector_type(2))) float v2f;
typedef __attribute__((ext_vector_type(8))) float v8f;

#define N_PTS   8192
#define D_DIM   256
#define BIGF    1e30f
#define MARGINF 0.1f

#define ROWS_PER_BLK 64
#define COLS_PER_IT  32
#define NSTEP        64            // K / 4
#define SUB_STRIDE   4128          // 64 steps * 64 dwords + 32-dword pad (bank decorrelation)

// ---------------- kernel 0: row sum-of-squares (wave per row) ----------------
__global__ __launch_bounds__(256) void vcl_sq_kernel(const float* __restrict__ q,
                                                     float* __restrict__ sq) {
    const int wave = threadIdx.x >> 5;
    const int lane = threadIdx.x & 31;
    const int row  = blockIdx.x * 8 + wave;
    const float* r = q + row * D_DIM;
    float s = 0.0f;
#pragma unroll
    for (int k = 0; k < D_DIM; k += 32) {
        float x = r[k + lane];
        s += x * x;
    }
#pragma unroll
    for (int m = 16; m >= 1; m >>= 1) s += __shfl_xor(s, m, 32);
    if (lane == 0) sq[row] = s;
}

// ---------------- kernel 1: fused WMMA gram + masked min/max + per-block loss ----------------
__global__ __launch_bounds__(256) void vcl_main_kernel(const float* __restrict__ q,
                                                       const int*   __restrict__ labels,
                                                       const int*   __restrict__ cams,
                                                       const float* __restrict__ sq,
                                                       float* __restrict__ partials) {
    __shared__ float Asw[4 * SUB_STRIDE];   // 64 rows, fragment-swizzled
    __shared__ float Bsw[2 * SUB_STRIDE];   // 32 cols, fragment-swizzled
    __shared__ int   rowLab[64], rowCam[64];
    __shared__ int   colLab[32], colCam[32];
    __shared__ float colSq[32];
    __shared__ float red[3][2][64];
    __shared__ float terms[64];

    const int tid     = threadIdx.x;
    const int lane    = tid & 31;
    const int wave    = tid >> 5;
    const int rowSub  = wave & 3;          // which 16-row sub-block
    const int colSub  = wave >> 2;         // which 16-col half of the 32-col tile
    const int halfsel = lane >> 4;
    const int cloc    = lane & 15;
    const int rowBase = blockIdx.x * ROWS_PER_BLK;

    // ---- stage A (64x256) into fragment-swizzled LDS; linear coalesced global reads ----
#pragma unroll
    for (int it = 0; it < 16; ++it) {
        int off = tid * 4 + it * 1024;                 // dword offset within 64x256 block
        int row = off >> 8;
        int k0  = off & 255;
        float4 f = *(const float4*)(q + rowBase * D_DIM + off);
        int sub = row >> 4, rr = row & 15, s = k0 >> 2;
        float* base = &Asw[sub * SUB_STRIDE + s * 64 + rr * 2];
        *(v2f*)(base)      = (v2f){f.x, f.y};          // K = k0,   k0+1  (half 0)
        *(v2f*)(base + 32) = (v2f){f.z, f.w};          // K = k0+2, k0+3  (half 1)
    }
    if (tid < 64) {
        rowLab[tid] = labels[rowBase + tid];
        rowCam[tid] = cams[rowBase + tid];
    }
    __syncthreads();

    // persistent per-lane row metadata (C-fragment layout: VGPR v -> row v + 8*half, col = lane&15)
    float sqR[8];
    int   labR[8], camR[8], growv[8];
    float maxI[8], minI[8], minE[8];
#pragma unroll
    for (int v = 0; v < 8; ++v) {
        int rl   = rowSub * 16 + v + 8 * halfsel;
        growv[v] = rowBase + rl;
        labR[v]  = rowLab[rl];
        camR[v]  = rowCam[rl];
        sqR[v]   = sq[rowBase + rl];
        maxI[v]  = -BIGF;
        minI[v]  =  BIGF;
        minE[v]  =  BIGF;
    }

    for (int cb = 0; cb < N_PTS; cb += COLS_PER_IT) {
        // ---- stage B (32x256): lane == B-row -> conflict-free fragment-order LDS writes ----
        {
            int r  = tid & 31;
            int sg = tid >> 5;
            const float* src = q + (cb + r) * D_DIM;
            int sub = r >> 4, rr = r & 15;
            float* dstbase = &Bsw[sub * SUB_STRIDE + rr * 2];
#pragma unroll
            for (int j = 0; j < 8; ++j) {
                int s = sg + j * 8;
                float4 f = *(const float4*)(src + s * 4);
                *(v2f*)(dstbase + s * 64)      = (v2f){f.x, f.y};
                *(v2f*)(dstbase + s * 64 + 32) = (v2f){f.z, f.w};
            }
            if (tid < 32) {
                colLab[tid] = labels[cb + tid];
                colCam[tid] = cams[cb + tid];
                colSq[tid]  = sq[cb + tid];
            }
        }
        __syncthreads();

        // ---- 16x16x256 gram tile: 64 chained f32 WMMAs, two accumulator chains ----
        const float* Ab = &Asw[rowSub * SUB_STRIDE + 2 * lane];
        const float* Bb = &Bsw[colSub * SUB_STRIDE + 2 * lane];
        v8f acc0 = {0.f,0.f,0.f,0.f,0.f,0.f,0.f,0.f};
        v8f acc1 = {0.f,0.f,0.f,0.f,0.f,0.f,0.f,0.f};
#pragma unroll 8
        for (int s = 0; s < NSTEP; s += 2) {
            v2f a0 = *(const v2f*)(Ab + s * 64);
            v2f b0 = *(const v2f*)(Bb + s * 64);
            v2f a1 = *(const v2f*)(Ab + (s + 1) * 64);
            v2f b1 = *(const v2f*)(Bb + (s + 1) * 64);
            acc0 = __builtin_amdgcn_wmma_f32_16x16x4_f32(false, a0, false, b0,
                                                         (short)0, acc0, false, false);
            acc1 = __builtin_amdgcn_wmma_f32_16x16x4_f32(false, a1, false, b1,
                                                         (short)0, acc1, false, false);
        }
        v8f acc = acc0 + acc1;

        // ---- fused distance + masked min/max update ----
        int   gcol = cb + colSub * 16 + cloc;
        int   labC = colLab[colSub * 16 + cloc];
        int   camC = colCam[colSub * 16 + cloc];
        float sqC  = colSq[colSub * 16 + cloc];
#pragma unroll
        for (int v = 0; v < 8; ++v) {
            float d2 = sqR[v] + sqC - 2.0f * acc[v];
            float d  = sqrtf(fmaxf(d2, 1e-12f));
            bool sid  = (labR[v] == labC);
            bool scam = (camR[v] == camC);
            bool self = (growv[v] == gcol);
            if (sid && scam && !self) {
                maxI[v] = fmaxf(maxI[v], d);
                minI[v] = fminf(minI[v], d);
            }
            if (sid && !scam) {
                minE[v] = fminf(minE[v], d);
            }
        }
        __syncthreads();   // before next B overwrite
    }

    // ---- reduce across the 16 column lanes of each half ----
#pragma unroll
    for (int v = 0; v < 8; ++v) {
#pragma unroll
        for (int m = 1; m <= 8; m <<= 1) {
            maxI[v] = fmaxf(maxI[v], __shfl_xor(maxI[v], m, 32));
            minI[v] = fminf(minI[v], __shfl_xor(minI[v], m, 32));
            minE[v] = fminf(minE[v], __shfl_xor(minE[v], m, 32));
        }
    }
    if (cloc == 0) {   // lanes 0 and 16 hold the reduced halves
#pragma unroll
        for (int v = 0; v < 8; ++v) {
            int rl = rowSub * 16 + v + 8 * halfsel;
            red[0][colSub][rl] = maxI[v];
            red[1][colSub][rl] = minI[v];
            red[2][colSub][rl] = minE[v];
        }
    }
    __syncthreads();

    // ---- per-row loss term (combine the two column-half waves) ----
    if (tid < 64) {
        float mx  = fmaxf(red[0][0][tid], red[0][1][tid]);
        float mnI = fminf(red[1][0][tid], red[1][1][tid]);
        float mnE = fminf(red[2][0][tid], red[2][1][tid]);
        bool hasI = mx  > -0.5f * BIGF;
        bool hasE = mnE <  0.5f * BIGF;
        float easyI = hasI ? mnI : 1.0f;
        float hardI = hasI ? mx  : 1.0f;
        float easyE = hasE ? mnE : hardI;
        terms[tid] = fmaxf(0.0f, MARGINF - (easyI - easyE));
    }
    __syncthreads();
    if (tid == 0) {
        float s = 0.0f;
        for (int i = 0; i < 64; ++i) s += terms[i];
        partials[blockIdx.x] = s;
    }
}

// ---------------- kernel 2: deterministic final reduction ----------------
__global__ void vcl_final_kernel(const float* __restrict__ partials, float* __restrict__ out) {
    if (threadIdx.x == 0) {
        float s = 0.0f;
        for (int i = 0; i < N_PTS / ROWS_PER_BLK; ++i) s += partials[i];
        out[0] = s / (float)N_PTS;
    }
}

extern "C" void kernel_launch(void* const* d_in, const int* in_sizes, int n_in,
                              void* d_out, int out_size, void* d_ws, size_t ws_size,
                              hipStream_t stream) {
    const float* q      = (const float*)d_in[0];
    const int*   labels = (const int*)d_in[1];
    const int*   cams   = (const int*)d_in[2];
    float* sq       = (float*)d_ws;          // N_PTS floats
    float* partials = sq + N_PTS;            // 128 floats
    float* out      = (float*)d_out;

    vcl_sq_kernel<<<N_PTS / 8, 256, 0, stream>>>(q, sq);
    vcl_main_kernel<<<N_PTS / ROWS_PER_BLK, 256, 0, stream>>>(q, labels, cams, sq, partials);
    vcl_final_kernel<<<1, 32, 0, stream>>>(partials, out);
}